// HetSynFSNNCell_59399397704145
// MI455X (gfx1250) — compile-verified
//
#include <hip/hip_runtime.h>
#include <math.h>

#define BLOCK 256

typedef float v4f __attribute__((ext_vector_type(4)));
// Match the builtin's parameter type exactly: gcc-style vector of 4 ints.
typedef int i4 __attribute__((vector_size(16)));
typedef __attribute__((address_space(1))) i4 gi4;   // global int4
typedef __attribute__((address_space(3))) i4 li4;   // LDS int4

// gfx1250 async global->LDS path (ASYNCcnt-tracked); guarded so the TU always
// compiles even if a toolchain lacks the builtins.
#if defined(__HIP_DEVICE_COMPILE__) && defined(__gfx1250__) && \
    __has_builtin(__builtin_amdgcn_global_load_async_to_lds_b128) && \
    __has_builtin(__builtin_amdgcn_s_wait_asynccnt)
#define USE_ASYNC_LDS 1
#else
#define USE_ASYNC_LDS 0
#endif

// cpol immediate: TH bits [2:0]; 1 = NT (non-temporal) — hIsyn is a 256MB
// read-once stream, keep it from evicting the L2-resident rho/w/x (~8MB).
#define CPOL_NT 1

__global__ __launch_bounds__(BLOCK) void hetsyn_fused_kernel(
    const float* __restrict__ x,      // (B, NI)
    const float* __restrict__ w,      // (NO, NI)
    const float* __restrict__ rho,    // (NO, NI)
    const float* __restrict__ hz,     // (B, NO)
    const float* __restrict__ hIsyn,  // (B, NO, NI)
    const float* __restrict__ hIr,    // (B, NO)
    float* __restrict__ out_z,        // (B, NO)
    float* __restrict__ out_Isyn,     // (B, NO, NI)
    float* __restrict__ out_Ir,       // (B, NO)
    int NO, int NI)
{
  const int tid = threadIdx.x;
  const int j   = blockIdx.x;              // output neuron
  const int b   = blockIdx.y;              // batch element
  const int row = b * NO + j;              // flat (b, j) row
  const size_t rowOff = (size_t)row * (size_t)NI;

  const float* __restrict__ hrow = hIsyn + rowOff;
  const float* __restrict__ rrow = rho + (size_t)j * NI;
  const float* __restrict__ wrow = w   + (size_t)j * NI;
  const float* __restrict__ xrow = x   + (size_t)b * NI;
  float* __restrict__ orow = out_Isyn + rowOff;

#if USE_ASYNC_LDS
  __shared__ float lds_h[BLOCK * 4];
#endif

  float acc = 0.0f;
  const int step = BLOCK * 4;
  for (int i0 = tid * 4; i0 + 3 < NI; i0 += step) {
#if USE_ASYNC_LDS
    // Kick off the zero-reuse hIsyn stream via async DMA into LDS (NT hint),
    // then load the L2-resident operands while it is in flight.
    __builtin_amdgcn_global_load_async_to_lds_b128(
        (gi4*)(hrow + i0),
        (li4*)(lds_h + tid * 4),
        0, CPOL_NT);
    v4f r4 = *(const v4f*)(rrow + i0);
    v4f w4 = *(const v4f*)(wrow + i0);
    v4f x4 = *(const v4f*)(xrow + i0);
    __builtin_amdgcn_s_wait_asynccnt(0);   // each lane reads only its own bytes
    v4f h4 = *(const v4f*)(lds_h + tid * 4);
#else
    v4f h4 = __builtin_nontemporal_load((const v4f*)(hrow + i0));
    v4f r4 = *(const v4f*)(rrow + i0);
    v4f w4 = *(const v4f*)(wrow + i0);
    v4f x4 = *(const v4f*)(xrow + i0);
#endif
    v4f isyn = r4 * h4 + w4 * x4;
    __builtin_nontemporal_store(isyn, (v4f*)(orow + i0));   // streaming store
    acc += (isyn.x + isyn.y) + (isyn.z + isyn.w);
  }
  // scalar tail for NI % 4 != 0 (not hit for NI=1024)
  for (int i = (NI & ~3) + tid; i < NI; i += BLOCK) {
    float isv = rrow[i] * hrow[i] + wrow[i] * xrow[i];
    orow[i] = isv;
    acc += isv;
  }

  // wave32 butterfly reduction
  #pragma unroll
  for (int off = 16; off > 0; off >>= 1)
    acc += __shfl_xor(acc, off, 32);

  __shared__ float wsum[BLOCK / 32];
  const int lane = tid & 31;
  const int wid  = tid >> 5;
  if (lane == 0) wsum[wid] = acc;
  __syncthreads();

  if (tid == 0) {
    float v_in = 0.0f;
    #pragma unroll
    for (int k = 0; k < BLOCK / 32; ++k) v_in += wsum[k];
    const float RHO_R = 0.95122942450071400910f;  // exp(-0.001/0.02)
    const float THR   = 1.0f;
    float Ir = RHO_R * hIr[row] + hz[row];
    float v  = v_in - THR * Ir;
    out_z[row]  = (v - THR >= 0.0f) ? 1.0f : 0.0f;
    out_Ir[row] = Ir;
  }
}

extern "C" void kernel_launch(void* const* d_in, const int* in_sizes, int n_in,
                              void* d_out, int out_size, void* d_ws, size_t ws_size,
                              hipStream_t stream) {
  const float* x     = (const float*)d_in[0];
  const float* w     = (const float*)d_in[1];
  const float* rho   = (const float*)d_in[2];
  const float* hz    = (const float*)d_in[3];
  const float* hIsyn = (const float*)d_in[4];
  const float* hIr   = (const float*)d_in[5];

  // B*NI = |x|, NO*NI = |w|, B*NO = |hz|  =>  NI = sqrt(|x|*|w|/|hz|)
  const double Sx = (double)in_sizes[0];
  const double Sw = (double)in_sizes[1];
  const double Sz = (double)in_sizes[3];
  const int NI = (int)llround(sqrt(Sx * Sw / Sz));
  const int B  = (int)llround(Sx / (double)NI);
  const int NO = (int)llround(Sw / (double)NI);

  // outputs concatenated flat in return order: z, Isyn, Ir
  float* out_z    = (float*)d_out;
  float* out_Isyn = out_z + (size_t)B * NO;
  float* out_Ir   = out_Isyn + (size_t)B * NO * NI;

  dim3 grid((unsigned)NO, (unsigned)B), block(BLOCK);
  hipLaunchKernelGGL(hetsyn_fused_kernel, grid, block, 0, stream,
                     x, w, rho, hz, hIsyn, hIr, out_z, out_Isyn, out_Ir, NO, NI);
}